// PointNetSetAbstractionMsg_55688545960048
// MI455X (gfx1250) — compile-verified
//
#include <hip/hip_runtime.h>

#define BATCH  16
#define NPTS   8192
#define NPOINT 512
#define OUTCH  320

typedef float v2f __attribute__((ext_vector_type(2)));
typedef float v8f __attribute__((ext_vector_type(8)));
typedef unsigned int v4u __attribute__((ext_vector_type(4)));
typedef int v4i __attribute__((ext_vector_type(4)));
typedef int v8i __attribute__((ext_vector_type(8)));

// ---------------------------------------------------------------------------
// 1. Farthest point sampling: one block per batch, min-dist array in LDS.
//    Also writes new_xyz to d_out (as [B,3,S]) and to ws (as [B*S,3]).
// ---------------------------------------------------------------------------
__global__ void fps_kernel(const float* __restrict__ xyz, int* __restrict__ cent,
                           float* __restrict__ newxyz, float* __restrict__ out_nx)
{
    __shared__ float dist[NPTS];
    __shared__ float redv[512];
    __shared__ int   redi[512];
    const int b = blockIdx.x, tid = threadIdx.x;
    const float* xb = xyz + (size_t)b * 3 * NPTS;
    for (int i = tid; i < NPTS; i += 512) dist[i] = 1e10f;
    __syncthreads();
    int far = 0;
    for (int it = 0; it < NPOINT; ++it) {
        const float cx = xb[far], cy = xb[NPTS + far], cz = xb[2 * NPTS + far];
        if (tid == 0) {
            cent[b * NPOINT + it] = far;
            float* nx = newxyz + ((size_t)b * NPOINT + it) * 3;
            nx[0] = cx; nx[1] = cy; nx[2] = cz;
            out_nx[(size_t)b * 3 * NPOINT + 0 * NPOINT + it] = cx;
            out_nx[(size_t)b * 3 * NPOINT + 1 * NPOINT + it] = cy;
            out_nx[(size_t)b * 3 * NPOINT + 2 * NPOINT + it] = cz;
        }
        float bv = -1.0f; int bi = 0;
        for (int i = tid; i < NPTS; i += 512) {
            const float dx = xb[i] - cx, dy = xb[NPTS + i] - cy, dz = xb[2 * NPTS + i] - cz;
            const float d = dx * dx + dy * dy + dz * dz;
            const float m = fminf(dist[i], d);
            dist[i] = m;
            if (m > bv) { bv = m; bi = i; }
        }
        redv[tid] = bv; redi[tid] = bi;
        __syncthreads();
        for (int off = 256; off > 0; off >>= 1) {
            if (tid < off && redv[tid + off] > redv[tid]) {
                redv[tid] = redv[tid + off]; redi[tid] = redi[tid + off];
            }
            __syncthreads();
        }
        far = redi[0];
        __syncthreads();
    }
}

// ---------------------------------------------------------------------------
// 2. Ball query + grouping: one wave32 per centroid. Order-preserving
//    compaction of "first K indices with d2 <= r2" via ballot/popcount.
//    Writes [points(3), xyz-center(3), 0, 0] rows (Cin padded 6 -> 8).
// ---------------------------------------------------------------------------
__global__ void group_kernel(const float* __restrict__ xyz, const float* __restrict__ pts,
                             const float* __restrict__ newxyz, float* __restrict__ feat,
                             float r2, int K)
{
    const int wave = threadIdx.x >> 5, lane = threadIdx.x & 31;
    const int s_lin = blockIdx.x * 8 + wave;
    const int b = s_lin >> 9;                          // S = 512
    const float* xb = xyz + (size_t)b * 3 * NPTS;
    const float* pb = pts + (size_t)b * 3 * NPTS;
    const float cx = newxyz[(size_t)s_lin * 3 + 0];
    const float cy = newxyz[(size_t)s_lin * 3 + 1];
    const float cz = newxyz[(size_t)s_lin * 3 + 2];

    int cnt = 0, firstIdx = -1;
    for (int base = 0; base < NPTS && cnt < K; base += 32) {
        const int j = base + lane;
        const float dx = xb[j] - cx, dy = xb[NPTS + j] - cy, dz = xb[2 * NPTS + j] - cz;
        const float d = dx * dx + dy * dy + dz * dz;
        const bool within = (d <= r2);
        const unsigned long long mask = __ballot(within);
        if (firstIdx < 0 && mask)
            firstIdx = base + (int)__builtin_ctzll(mask);
        const int pre = __popcll(mask & ((1ull << lane) - 1ull));
        const int pos = cnt + pre;
        if (within && pos < K) {
            float* f = feat + ((size_t)s_lin * K + pos) * 8;
            f[0] = pb[j]; f[1] = pb[NPTS + j]; f[2] = pb[2 * NPTS + j];
            f[3] = dx;    f[4] = dy;           f[5] = dz;
            f[6] = 0.f;   f[7] = 0.f;
        }
        cnt += (int)__popcll(mask);
    }
    if (cnt < K && firstIdx >= 0) {                    // pad with first neighbor
        const int j = firstIdx;
        const float p0 = pb[j], p1 = pb[NPTS + j], p2 = pb[2 * NPTS + j];
        const float g0 = xb[j] - cx, g1 = xb[NPTS + j] - cy, g2 = xb[2 * NPTS + j] - cz;
        for (int p = cnt + lane; p < K; p += 32) {
            float* f = feat + ((size_t)s_lin * K + p) * 8;
            f[0] = p0; f[1] = p1; f[2] = p2; f[3] = g0; f[4] = g1; f[5] = g2;
            f[6] = 0.f; f[7] = 0.f;
        }
    }
}

// ---------------------------------------------------------------------------
// 3. GEMM + bias via V_WMMA_F32_16X16X4_F32 (f32-exact on matrix pipes).
//    Z[M, Cout] = X[M, Cin] * W[Cout, Cin]^T + bias.
//    Weights staged into LDS by the Tensor Data Mover: one wave issues a
//    single tensor_load_to_lds whose tile is [Cout x kloop] while the tensor
//    is [Cout x ldw] -- the TDM's OOB-returns-zero rule performs the k >= ldw
//    zero padding in hardware. Hot loop is guard-free:
//    per k-step = 2 global_b64 (A) + NT ds_b64 (B) + 2*NT wmma.
// ---------------------------------------------------------------------------
template <int NT>
__global__ void __launch_bounds__(256)
gemm_kernel(const float* __restrict__ X, int ldx,
            const float* __restrict__ W, int ldw, int kloop,
            const float* __restrict__ bias,
            float* __restrict__ Z, int Mtot)
{
    __shared__ float w_lds[128 * 96];                  // max Cout*kloop = 48 KB
    const int Cout = NT * 16;

    if (threadIdx.x < 32) {                            // wave 0 drives the TDM
        const unsigned ldsOff = (unsigned)(unsigned long long)(void*)w_lds;
        const unsigned long long ga = (unsigned long long)W;
        v4u g0;
        g0[0] = 1u;                                    // count=1, user descriptor
        g0[1] = ldsOff;                                // D#.lds_addr
        g0[2] = (unsigned)(ga & 0xFFFFFFFFull);        // D#.global_addr[31:0]
        g0[3] = (unsigned)((ga >> 32) & 0x1FFFFFFull)  // D#.global_addr[56:32]
              | (2u << 30);                            // type = 2 (image)
        v8i g1;
        g1[0] = (int)(2u << 16);                       // data_size = 4 bytes
        g1[1] = (int)(((unsigned)ldw & 0xFFFFu) << 16);        // tensor_dim0 lo
        g1[2] = (int)((((unsigned)ldw >> 16) & 0xFFFFu)        // tensor_dim0 hi
              | (((unsigned)Cout & 0xFFFFu) << 16));           // tensor_dim1 lo
        g1[3] = (int)((((unsigned)Cout >> 16) & 0xFFFFu)       // tensor_dim1 hi
              | (((unsigned)kloop & 0xFFFFu) << 16));          // tile_dim0 = kloop
        g1[4] = (int)((unsigned)Cout & 0xFFFFu);               // tile_dim1 = Cout, tile_dim2 = 0
        g1[5] = (int)(unsigned)ldw;                            // tensor_dim0_stride[31:0] = ldw
        g1[6] = 0;                                             // stride0 hi / stride1 lo
        g1[7] = 0;
        const v4i g2 = {0, 0, 0, 0};
        const v4i g3 = {0, 0, 0, 0};
        const v8i g4 = {0, 0, 0, 0, 0, 0, 0, 0};
        __builtin_amdgcn_tensor_load_to_lds(g0, g1, g2, g3, g4, 0);
        __builtin_amdgcn_s_wait_tensorcnt(0);
    }
    __syncthreads();

    const int wid  = (int)((blockIdx.x * blockDim.x + threadIdx.x) >> 5);
    const int lane = threadIdx.x & 31;
    const int m0 = wid * 32;                           // two 16-row tiles
    if (m0 >= Mtot) return;
    const int r  = lane & 15;
    const int kh = lane >> 4;                          // 0 or 1: k-pair {0,1} vs {2,3}
    const v8f zero = {0.f, 0.f, 0.f, 0.f, 0.f, 0.f, 0.f, 0.f};
    v8f acc0[NT], acc1[NT];
#pragma unroll
    for (int t = 0; t < NT; ++t) { acc0[t] = zero; acc1[t] = zero; }

    const float* xrow0 = X + (size_t)(m0 + r) * ldx;
    const float* xrow1 = X + (size_t)(m0 + 16 + r) * ldx;
    for (int k = 0; k < kloop; k += 4) {
        const int ka = k + kh * 2;
        const v2f a0 = *(const v2f*)(xrow0 + ka);
        const v2f a1 = *(const v2f*)(xrow1 + ka);
#pragma unroll
        for (int t = 0; t < NT; ++t) {
            const int col = t * 16 + r;
            const v2f bb = *(const v2f*)&w_lds[col * kloop + ka];
            acc0[t] = __builtin_amdgcn_wmma_f32_16x16x4_f32(
                false, a0, false, bb, (short)0, acc0[t], false, false);
            acc1[t] = __builtin_amdgcn_wmma_f32_16x16x4_f32(
                false, a1, false, bb, (short)0, acc1[t], false, false);
        }
    }
#pragma unroll
    for (int t = 0; t < NT; ++t) {
        const int col = t * 16 + r;
        const float bv = bias[col];
#pragma unroll
        for (int i = 0; i < 8; ++i) {                  // C/D layout: VGPR i -> M = i (+8 hi half)
            Z[(size_t)(m0 + i + kh * 8) * Cout + col]      = acc0[t][i] + bv;
            Z[(size_t)(m0 + 16 + i + kh * 8) * Cout + col] = acc1[t][i] + bv;
        }
    }
}

// ---------------------------------------------------------------------------
// 4. BatchNorm (batch statistics over all B*S*K) + ReLU, and max-over-K.
// ---------------------------------------------------------------------------
__global__ void zero_stats_kernel(float* stats)
{
    if (threadIdx.x < 256) stats[threadIdx.x] = 0.f;
}

__global__ void bn_stats_kernel(const float* __restrict__ Z, int Cout, int Mtot,
                                float* __restrict__ stats)
{
    const int c = threadIdx.x;
    if (c >= Cout) return;
    float s = 0.f, q = 0.f;
    for (int row = blockIdx.x; row < Mtot; row += gridDim.x) {
        const float v = Z[(size_t)row * Cout + c];
        s += v; q += v * v;
    }
    atomicAdd(&stats[c], s);
    atomicAdd(&stats[128 + c], q);
}

__global__ void bn_finalize_kernel(float* stats, const float* __restrict__ gamma,
                                   const float* __restrict__ beta, int Cout, float invM)
{
    const int c = threadIdx.x;
    if (c >= Cout) return;
    const float mu  = stats[c] * invM;
    const float var = stats[128 + c] * invM - mu * mu;
    const float sc  = gamma[c] * rsqrtf(var + 1e-5f);
    stats[c]       = sc;
    stats[128 + c] = beta[c] - mu * sc;
}

__global__ void bn_relu_kernel(const float4* __restrict__ Z, const float* __restrict__ stats,
                               float4* __restrict__ O, int Cout, long total4)
{
    const long i = (long)blockIdx.x * blockDim.x + threadIdx.x;
    if (i >= total4) return;
    const int c = (int)((i * 4) % Cout);
    float4 v = Z[i];
    v.x = fmaxf(v.x * stats[c + 0] + stats[128 + c + 0], 0.f);
    v.y = fmaxf(v.y * stats[c + 1] + stats[128 + c + 1], 0.f);
    v.z = fmaxf(v.z * stats[c + 2] + stats[128 + c + 2], 0.f);
    v.w = fmaxf(v.w * stats[c + 3] + stats[128 + c + 3], 0.f);
    O[i] = v;
}

__global__ void maxpool_kernel(const float* __restrict__ F, int Cout, int K,
                               float* __restrict__ out, int choff)
{
    const int i = blockIdx.x * blockDim.x + threadIdx.x;   // over B*S*Cout/4
    if (i * 4 >= BATCH * NPOINT * Cout) return;
    const int c = (i * 4) % Cout;
    const int s_lin = (i * 4) / Cout;
    const size_t m0 = (size_t)s_lin * K;
    float4 mx = make_float4(-1e30f, -1e30f, -1e30f, -1e30f);
    for (int k = 0; k < K; ++k) {
        const float4 v = *(const float4*)&F[(m0 + k) * Cout + c];
        mx.x = fmaxf(mx.x, v.x); mx.y = fmaxf(mx.y, v.y);
        mx.z = fmaxf(mx.z, v.z); mx.w = fmaxf(mx.w, v.w);
    }
    const int b = s_lin >> 9, s = s_lin & 511;
    float* o = out + (size_t)b * OUTCH * NPOINT + (size_t)(choff + c) * NPOINT + s;
    o[0 * NPOINT] = mx.x; o[1 * NPOINT] = mx.y; o[2 * NPOINT] = mx.z; o[3 * NPOINT] = mx.w;
}

// ---------------------------------------------------------------------------
// Driver
// ---------------------------------------------------------------------------
extern "C" void kernel_launch(void* const* d_in, const int* in_sizes, int n_in,
                              void* d_out, int out_size, void* d_ws, size_t ws_size,
                              hipStream_t stream)
{
    (void)in_sizes; (void)n_in; (void)out_size; (void)ws_size;
    const float* xyz = (const float*)d_in[0];
    const float* pts = (const float*)d_in[1];

    char*  ws     = (char*)d_ws;
    int*   cent   = (int*)ws;                                 // 16*512 ints
    float* newxyz = (float*)(ws + 32768);                     // 16*512*3 f32
    float* stats  = (float*)(ws + 32768 + 98304);             // 256 f32
    float* featA  = (float*)(ws + 132096);                    // up to 1M x 128 f32
    float* featZ  = featA + (size_t)134217728;                // second 512 MB buffer

    float* out_nx = (float*)d_out;                            // [B,3,S]
    float* out_np = (float*)d_out + (size_t)BATCH * 3 * NPOINT; // [B,320,S]

    fps_kernel<<<BATCH, 512, 0, stream>>>(xyz, cent, newxyz, out_nx);

    const int   Ks[3]     = {16, 32, 128};
    const float R2s[3]    = {0.01f, 0.04f, 0.16f};
    const int   mlp[3][3] = {{32, 32, 64}, {64, 64, 128}, {64, 96, 128}};
    const int   choffs[3] = {0, 64, 192};

    for (int sc = 0; sc < 3; ++sc) {
        const int K    = Ks[sc];
        const int Mtot = BATCH * NPOINT * K;

        group_kernel<<<BATCH * NPOINT / 8, 256, 0, stream>>>(xyz, pts, newxyz,
                                                             featA, R2s[sc], K);

        int cin = 6, ldx = 8, kloop = 8;                      // first layer padded 6 -> 8
        for (int l = 0; l < 3; ++l) {
            const int cout = mlp[sc][l];
            const int pi = 2 + (sc * 3 + l) * 4;
            const float* W  = (const float*)d_in[pi + 0];
            const float* bb = (const float*)d_in[pi + 1];
            const float* gm = (const float*)d_in[pi + 2];
            const float* bt = (const float*)d_in[pi + 3];

            const int nblk = Mtot / 256;                      // Mtot/32 waves, 8 waves/block
            switch (cout / 16) {
            case 2: gemm_kernel<2><<<nblk, 256, 0, stream>>>(featA, ldx, W, cin, kloop, bb, featZ, Mtot); break;
            case 4: gemm_kernel<4><<<nblk, 256, 0, stream>>>(featA, ldx, W, cin, kloop, bb, featZ, Mtot); break;
            case 6: gemm_kernel<6><<<nblk, 256, 0, stream>>>(featA, ldx, W, cin, kloop, bb, featZ, Mtot); break;
            default: gemm_kernel<8><<<nblk, 256, 0, stream>>>(featA, ldx, W, cin, kloop, bb, featZ, Mtot); break;
            }

            zero_stats_kernel<<<1, 256, 0, stream>>>(stats);
            bn_stats_kernel<<<1024, 128, 0, stream>>>(featZ, cout, Mtot, stats);
            bn_finalize_kernel<<<1, 128, 0, stream>>>(stats, gm, bt, cout, 1.0f / (float)Mtot);

            const long total4 = (long)Mtot * cout / 4;
            bn_relu_kernel<<<(int)((total4 + 255) / 256), 256, 0, stream>>>(
                (const float4*)featZ, stats, (float4*)featA, cout, total4);

            cin = cout; ldx = cout; kloop = cout;
        }

        const int lastC = mlp[sc][2];
        const int tot4 = BATCH * NPOINT * lastC / 4;
        maxpool_kernel<<<(tot4 + 255) / 256, 256, 0, stream>>>(featA, lastC, K, out_np, choffs[sc]);
    }
}